// TransformerModel_27530740368027
// MI455X (gfx1250) — compile-verified
//
#include <hip/hip_runtime.h>

// Dims from reference
#define T_STEPS 48
#define B_DIM   64
#define V_DIM   256
#define C_DIM   512
#define A_DIM   256
#define S_DIM   160

typedef __attribute__((ext_vector_type(2))) float v2f;
typedef __attribute__((ext_vector_type(8))) float v8f;

// D = A(16x4) * B(4x16) + C, fp32 WMMA (CDNA5 V_WMMA_F32_16X16X4_F32)
__device__ __forceinline__ v8f wmma_k4(v2f a, v2f b, v8f c) {
  return __builtin_amdgcn_wmma_f32_16x16x4_f32(false, a, false, b, (short)0, c,
                                               false, false);
}

__device__ __forceinline__ float wave_sum(float v) {
#pragma unroll
  for (int m = 16; m >= 1; m >>= 1) v += __shfl_xor(v, m, 32);
  return v;
}
__device__ __forceinline__ float wave_max(float v) {
#pragma unroll
  for (int m = 16; m >= 1; m >>= 1) v = fmaxf(v, __shfl_xor(v, m, 32));
  return v;
}

// Y[16][256] (LDS) += A[16][lda] * W[K][256].   All 32 waves participate:
// wave w -> N-tile (w&15), K-half (w>>4). Y must be pre-initialized (bias)
// and a __syncthreads() issued before/after by the caller. Accumulation into
// Y uses LDS float atomics (ds_add_f32) so the two K-half waves can combine
// without an extra barrier.
__device__ __forceinline__ void gemm16_accum(const float* A, int lda,
                                             const float* __restrict__ W,
                                             float* Yl, int K) {
  const int lane = threadIdx.x & 31;
  const int wave = threadIdx.x >> 5;
  const int n = wave & 15, kh = wave >> 4;
  const int m = lane & 15, half = lane >> 4;
  const int col = n * 16 + m;
  const float* Arow = A + m * lda;
  v8f acc = {0.f, 0.f, 0.f, 0.f, 0.f, 0.f, 0.f, 0.f};
  const int k0 = kh * (K >> 1), k1 = k0 + (K >> 1);
  for (int k = k0; k < k1; k += 4) {
    const int ka = k + 2 * half;                 // ISA A-layout: halves hold K+0/1 vs K+2/3
    v2f a; a.x = Arow[ka]; a.y = Arow[ka + 1];
    v2f b; b.x = W[(size_t)ka * 256 + col];      // B-layout: N striped on lanes, K by half
    b.y = W[(size_t)(ka + 1) * 256 + col];
    acc = wmma_k4(a, b, acc);
  }
#pragma unroll
  for (int r = 0; r < 8; ++r) {                  // D: vgpr r -> M=r (+8 for hi half), N=col
    const int row = r + 8 * half;
    atomicAdd(&Yl[row * 256 + col], acc[r]);
  }
}

// ---------------- Kernel 1: img_k = img_features @ Wk_i + bk_i ----------------
// grid = (S/16)*B = 640 blocks, 256 threads (8 waves). Block -> (b, s-tile);
// each wave owns two 16x16 N-tiles, full K=512.
__global__ __launch_bounds__(256) void imgk_kernel(
    const float* __restrict__ imgf, const float* __restrict__ Wki,
    const float* __restrict__ bki, float* __restrict__ imgk) {
  const int lane = threadIdx.x & 31, wave = threadIdx.x >> 5;
  const int b  = blockIdx.x & (B_DIM - 1);
  const int s0 = (blockIdx.x >> 6) * 16;
  const int m = lane & 15, half = lane >> 4;
  const float* Arow = imgf + ((size_t)(s0 + m) * B_DIM + b) * C_DIM;
#pragma unroll
  for (int it = 0; it < 2; ++it) {
    const int n = wave * 2 + it;
    const int col = n * 16 + m;
    v8f acc = {0.f, 0.f, 0.f, 0.f, 0.f, 0.f, 0.f, 0.f};
    for (int k = 0; k < C_DIM; k += 4) {
      const int ka = k + 2 * half;
      v2f a; a.x = Arow[ka]; a.y = Arow[ka + 1];
      v2f bb; bb.x = Wki[(size_t)ka * 256 + col];
      bb.y = Wki[(size_t)(ka + 1) * 256 + col];
      acc = wmma_k4(a, bb, acc);
    }
    const float bias = bki[col];
#pragma unroll
    for (int r = 0; r < 8; ++r) {
      const int row = r + 8 * half;
      imgk[((size_t)(s0 + row) * B_DIM + b) * A_DIM + col] = acc[r] + bias;
    }
  }
}

// ---------------- Kernel 2: persistent decode ----------------
// 4 blocks x 1024 threads; block owns 16 batch rows, runs all 48 steps with
// only intra-WGP barriers. All intermediates live in LDS (48KB).
__global__ __launch_bounds__(1024) void decode_kernel(
    const float* __restrict__ imgf, const float* __restrict__ start,
    const float* __restrict__ Wq_t, const float* __restrict__ bq_t,
    const float* __restrict__ Wk_t, const float* __restrict__ bk_t,
    const float* __restrict__ v_t,
    const float* __restrict__ Wq_i, const float* __restrict__ bq_i,
    const float* __restrict__ v_i,
    const float* __restrict__ Wc, const float* __restrict__ bc,
    const float* __restrict__ imgk,
    float* __restrict__ hist, float* __restrict__ proj,
    float* __restrict__ dout) {
  __shared__ float sA[16 * 256];                 // q / ctx_text / out accumulator
  __shared__ float sB[16 * 256];                 // qW / cq / ctx-chunk / proj
  __shared__ float sWT[(T_STEPS + 1) * 16];      // text attn logits/weights
  __shared__ float sWI[S_DIM * 16];              // image attn logits/weights
  __shared__ float sVt[256];
  __shared__ float sVi[256];

  const int tid = threadIdx.x;
  const int lane = tid & 31, wave = tid >> 5;
  const int b0 = blockIdx.x * 16;

  for (int e = tid; e < 256; e += 1024) { sVt[e] = v_t[e]; sVi[e] = v_i[e]; }
  // hist[0] = start; sA = start rows
  for (int e = tid; e < 4096; e += 1024) {
    const int bl = e >> 8, v = e & 255;
    const float x = start[(size_t)(b0 + bl) * V_DIM + v];
    sA[e] = x;
    hist[(size_t)(b0 + bl) * V_DIM + v] = x;
  }
  // proj[0] = start @ Wk_t + bk_t
  for (int e = tid; e < 4096; e += 1024) sB[e] = bk_t[e & 255];
  __syncthreads();
  gemm16_accum(sA, 256, Wk_t, sB, 256);
  __syncthreads();
  for (int e = tid; e < 4096; e += 1024)
    proj[(size_t)(b0 + (e >> 8)) * V_DIM + (e & 255)] = sB[e];
  __syncthreads();

  for (int t = 0; t < T_STEPS; ++t) {
    const int nH = t + 1;
    // ---- qW = q @ Wq_t + bq_t  (q in sA) -> sB
    for (int e = tid; e < 4096; e += 1024) sB[e] = bq_t[e & 255];
    __syncthreads();
    gemm16_accum(sA, 256, Wq_t, sB, 256);
    __syncthreads();
    // ---- text attention logits: e_t[i,b] = v_t . tanh(proj[i,b,:] + qW[b,:])
    for (int task = wave; task < nH * 16; task += 32) {
      const int i = task >> 4, bl = task & 15;
      const float* pr = proj + ((size_t)i * B_DIM + b0 + bl) * V_DIM;
      const float* qw = sB + bl * 256;
      float s = 0.f;
#pragma unroll
      for (int j = 0; j < 8; ++j) {
        const int v = lane + 32 * j;
        s += sVt[v] * tanhf(pr[v] + qw[v]);
      }
      s = wave_sum(s);
      if (lane == 0) sWT[i * 16 + bl] = s;
    }
    __syncthreads();
    // ---- softmax over history (one wave per batch row)
    if (wave < 16) {
      const int bl = wave;
      float mx = -3.4e38f;
      for (int i = lane; i < nH; i += 32) mx = fmaxf(mx, sWT[i * 16 + bl]);
      mx = wave_max(mx);
      float sum = 0.f;
      for (int i = lane; i < nH; i += 32) {
        const float ex = __expf(sWT[i * 16 + bl] - mx);
        sWT[i * 16 + bl] = ex; sum += ex;
      }
      sum = wave_sum(sum);
      const float inv = __builtin_amdgcn_rcpf(sum);
      for (int i = lane; i < nH; i += 32) sWT[i * 16 + bl] *= inv;
    }
    __syncthreads();
    // ---- ctx_text[b,v] = sum_i w[i,b]*hist[i,b,v] -> sA
    for (int e = tid; e < 4096; e += 1024) {
      const int bl = e >> 8, v = e & 255;
      const float* hb = hist + ((size_t)(b0 + bl)) * V_DIM + v;
      float s = 0.f;
#pragma unroll 4
      for (int i = 0; i < nH; ++i)
        s += sWT[i * 16 + bl] * hb[(size_t)i * B_DIM * V_DIM];
      sA[e] = s;
    }
    __syncthreads();
    // ---- cq = ctx_text @ Wq_i + bq_i -> sB
    for (int e = tid; e < 4096; e += 1024) sB[e] = bq_i[e & 255];
    __syncthreads();
    gemm16_accum(sA, 256, Wq_i, sB, 256);
    __syncthreads();
    // ---- image attention logits: e_i[s,b] = v_i . tanh(img_k[s,b,:] + cq[b,:])
    for (int task = wave; task < S_DIM * 16; task += 32) {
      const int s = task >> 4, bl = task & 15;
      const float* ik = imgk + ((size_t)s * B_DIM + b0 + bl) * A_DIM;
      const float* cq = sB + bl * 256;
      float acc = 0.f;
#pragma unroll
      for (int j = 0; j < 8; ++j) {
        const int a = lane + 32 * j;
        acc += sVi[a] * tanhf(ik[a] + cq[a]);
      }
      acc = wave_sum(acc);
      if (lane == 0) sWI[s * 16 + bl] = acc;
    }
    __syncthreads();
    // ---- softmax over S; concurrently bias-init out accumulator sA = bc
    if (wave < 16) {
      const int bl = wave;
      float mx = -3.4e38f;
      for (int s = lane; s < S_DIM; s += 32) mx = fmaxf(mx, sWI[s * 16 + bl]);
      mx = wave_max(mx);
      float sum = 0.f;
      for (int s = lane; s < S_DIM; s += 32) {
        const float ex = __expf(sWI[s * 16 + bl] - mx);
        sWI[s * 16 + bl] = ex; sum += ex;
      }
      sum = wave_sum(sum);
      const float inv = __builtin_amdgcn_rcpf(sum);
      for (int s = lane; s < S_DIM; s += 32) sWI[s * 16 + bl] *= inv;
    }
    for (int e = tid; e < 4096; e += 1024) sA[e] = bc[e & 255];
    __syncthreads();
    // ---- ctx (16x512) in two 256-chunks, each fed straight into the out-GEMM
    for (int h = 0; h < 2; ++h) {
      for (int e = tid; e < 4096; e += 1024) {
        const int bl = e >> 8, c = (e & 255) + h * 256;
        const float* base = imgf + (size_t)(b0 + bl) * C_DIM + c;
        float s = 0.f;
#pragma unroll 4
        for (int s0 = 0; s0 < S_DIM; ++s0) {
          if ((s0 & 15) == 0 && s0 + 16 < S_DIM)
            __builtin_prefetch(base + (size_t)(s0 + 16) * B_DIM * C_DIM, 0, 1);
          s += sWI[s0 * 16 + bl] * base[(size_t)s0 * B_DIM * C_DIM];
        }
        sB[e] = s;
      }
      __syncthreads();
      gemm16_accum(sB, 256, Wc + (size_t)h * 256 * 256, sA, 256);
      __syncthreads();
    }
    // ---- out = sA: emit to hist[t+1] and to d_out[t]
    for (int e = tid; e < 4096; e += 1024) {
      const int bl = e >> 8, v = e & 255;
      const float val = sA[e];
      hist[((size_t)(t + 1) * B_DIM + b0 + bl) * V_DIM + v] = val;
      dout[((size_t)t * B_DIM + b0 + bl) * V_DIM + v] = val;
    }
    // ---- proj[t+1] = out @ Wk_t + bk_t
    for (int e = tid; e < 4096; e += 1024) sB[e] = bk_t[e & 255];
    __syncthreads();
    gemm16_accum(sA, 256, Wk_t, sB, 256);
    __syncthreads();
    for (int e = tid; e < 4096; e += 1024)
      proj[((size_t)(t + 1) * B_DIM + b0 + (e >> 8)) * V_DIM + (e & 255)] = sB[e];
    __syncthreads();
    // loop invariant: sA holds q for step t+1
  }
}

extern "C" void kernel_launch(void* const* d_in, const int* in_sizes, int n_in,
                              void* d_out, int out_size, void* d_ws,
                              size_t ws_size, hipStream_t stream) {
  (void)in_sizes; (void)n_in; (void)out_size; (void)ws_size;
  const float* imgf  = (const float*)d_in[0];
  /* d_in[1] targets_onehot unused by the recurrence */
  const float* start = (const float*)d_in[2];
  const float* Wq_t  = (const float*)d_in[3];
  const float* bq_t  = (const float*)d_in[4];
  const float* Wk_t  = (const float*)d_in[5];
  const float* bk_t  = (const float*)d_in[6];
  const float* v_t   = (const float*)d_in[7];
  const float* Wq_i  = (const float*)d_in[8];
  const float* bq_i  = (const float*)d_in[9];
  const float* Wk_i  = (const float*)d_in[10];
  const float* bk_i  = (const float*)d_in[11];
  const float* v_i   = (const float*)d_in[12];
  const float* Wc    = (const float*)d_in[13];
  const float* bc    = (const float*)d_in[14];
  float* out = (float*)d_out;

  // workspace: img_k (10MB) + outputs history (3.2MB) + proj history (3.2MB)
  float* w_imgk = (float*)d_ws;
  float* w_hist = w_imgk + (size_t)S_DIM * B_DIM * A_DIM;
  float* w_proj = w_hist + (size_t)(T_STEPS + 1) * B_DIM * V_DIM;

  hipLaunchKernelGGL(imgk_kernel, dim3((S_DIM / 16) * B_DIM), dim3(256), 0,
                     stream, imgf, Wk_i, bk_i, w_imgk);
  hipLaunchKernelGGL(decode_kernel, dim3(B_DIM / 16), dim3(1024), 0, stream,
                     imgf, start, Wq_t, bq_t, Wk_t, bk_t, v_t, Wq_i, bq_i, v_i,
                     Wc, bc, w_imgk, w_hist, w_proj, out);
}